// ModelNew_80908593923259
// MI455X (gfx1250) — compile-verified
//
#include <hip/hip_runtime.h>
#include <hip/hip_bf16.h>

typedef float v2f __attribute__((ext_vector_type(2)));
typedef float v8f __attribute__((ext_vector_type(8)));
typedef int   v4i __attribute__((ext_vector_type(4)));

#define KTOT 216            // 8 * 3 * 3 * 3
#define NBLK_PER_B 105      // 7 pd * 15 ph
#define POOL_CNT 1575.0f    // 7*15*15

#define GLOBAL_AS __attribute__((address_space(1)))
#define LDS_AS    __attribute__((address_space(3)))

// Fused implicit-GEMM conv3d (f32 WMMA 16x16x4) + 2x2x2 maxpool + partial sum.
// Block = (ph, pd, b). 8 waves, each computes one 16ch x 16w conv tile at a
// distinct (d in {2pd,2pd+1}, h in {2ph,2ph+1}, w-half) via 54 chained WMMAs.
__global__ __launch_bounds__(256) void conv3d_wmma_pool_kernel(
    const float* __restrict__ x,      // [128,8,16,32,32]
    const float* __restrict__ wgt,    // [16,8,3,3,3] -> [16][216]
    float* __restrict__ partial,      // [128*105] or nullptr
    float* __restrict__ accum)        // [128] (atomic path) or nullptr
{
    __shared__ float sW[16 * KTOT];   // 13824 B  weight matrix A (16 x 216)
    __shared__ float sX[4104];        // input slab 8ci x 4d x 4h x 32w (+pad for w overrun)
    __shared__ int   sOfs[KTOT];      // k -> LDS BYTE offset (ci*512+kd*128+kh*32+kw)*4
    __shared__ float sScr[8][8][32];  // per-wave D tiles
    __shared__ float sSum[8];

    const int tid = threadIdx.x;
    const int ph = blockIdx.x;        // 0..14
    const int pd = blockIdx.y;        // 0..6
    const int b  = blockIdx.z;        // 0..127

    // --- stage weights (A matrix, row-major [co][k]) ---
    for (int i = tid; i < 16 * KTOT; i += 256) sW[i] = wgt[i];

    // --- k -> (ci,kd,kh,kw) LDS byte-offset table ---
    if (tid < KTOT) {
        int ci = tid / 27, r = tid % 27;
        int kd = r / 9, kh = (r % 9) / 3, kw = r % 3;
        sOfs[tid] = (ci * 512 + kd * 128 + kh * 32 + kw) * 4;
    }

    // --- stage input slab: x[b, ci, 2pd+0..3, 2ph+0..3, 0..31] (128 rows of 32f) ---
    {
        const float* xb = x + (size_t)b * (8 * 16 * 32 * 32);
#if defined(__gfx1250__) && __has_builtin(__builtin_amdgcn_global_load_async_to_lds_b128)
        // CDNA5 async global->LDS copy (no VGPR round-trip, tracked by ASYNCcnt)
        for (int i = tid; i < 1024; i += 256) {          // 1024 x 16B
            int row = i >> 3;                            // 0..127
            int q   = i & 7;
            int ci = row >> 4, dd = (row >> 2) & 3, hh = row & 3;
            const float* src = xb + (size_t)ci * 16384 + (size_t)(2 * pd + dd) * 1024 +
                               (size_t)(2 * ph + hh) * 32 + q * 4;
            __builtin_amdgcn_global_load_async_to_lds_b128(
                (GLOBAL_AS v4i*)src, (LDS_AS v4i*)(sX + i * 4), 0, 0);
        }
#if __has_builtin(__builtin_amdgcn_s_wait_asynccnt)
        __builtin_amdgcn_s_wait_asynccnt(0);
#else
        asm volatile("s_wait_asynccnt 0x0" ::: "memory");
#endif
#else
        for (int i = tid; i < 1024; i += 256) {          // 1024 float4
            int row = i >> 3;                            // 0..127
            int q   = i & 7;
            int ci = row >> 4, dd = (row >> 2) & 3, hh = row & 3;
            const float4* src = reinterpret_cast<const float4*>(
                xb + (size_t)ci * 16384 + (size_t)(2 * pd + dd) * 1024 +
                (size_t)(2 * ph + hh) * 32) + q;
            reinterpret_cast<float4*>(sX)[i] = *src;
        }
#endif
    }
    __syncthreads();

    const int wv    = tid >> 5;       // wave 0..7
    const int lane  = tid & 31;
    const int half  = lane >> 4;      // K sub-block select (ISA 16x4 layout)
    const int mn    = lane & 15;      // M for A-frag, N for B-frag
    const int dd    = wv & 1;         // d = 2pd + dd
    const int hh    = (wv >> 1) & 1;  // h = 2ph + hh
    const int whalf = wv >> 2;        // w0 = 16*whalf
    // per-lane base, in bytes, into sX
    const char* bBase = reinterpret_cast<const char*>(sX) +
                        (dd * 128 + hh * 32 + whalf * 16 + mn) * 4;
    const float* aRow = sW + mn * KTOT;

    v8f cacc = {};
    for (int k0 = 0; k0 < KTOT; k0 += 4) {
        const int ka = k0 + 2 * half;
        // A fragment: lanes 0-15 hold K=k0,k0+1; lanes 16-31 hold K=k0+2,k0+3
        v2f a;
        a.x = aRow[ka];
        a.y = aRow[ka + 1];
        // B fragment: same K split, N = lane&15 (column = conv w position).
        // Byte offsets; w-overrun columns read stale LDS but land only in
        // masked D columns (WMMA columns are independent).
        v2f bb;
        bb.x = *reinterpret_cast<const float*>(bBase + sOfs[ka]);
        bb.y = *reinterpret_cast<const float*>(bBase + sOfs[ka + 1]);
        // D = A*B + C  (v_wmma_f32_16x16x4_f32)
        cacc = __builtin_amdgcn_wmma_f32_16x16x4_f32(
            false, a, false, bb, (short)0, cacc, false, false);
    }

    // --- spill D tiles: D layout M = r + 8*(lane>=16), N = lane&15 ---
#pragma unroll
    for (int r = 0; r < 8; ++r) sScr[wv][r][lane] = cacc[r];
    __syncthreads();

    // --- 2x2x2 maxpool + sum over (channel, pooled-w): 16ch * 15pw = 240 vals ---
    float val = 0.0f;
    if (tid < 240) {
        int cch = tid / 15, pw = tid % 15;
        int wh = pw >> 3;                  // pw 0..7 -> half 0, pw 8..14 -> half 1
        int n0 = 2 * pw - wh * 16;         // even column in tile
        int r  = cch & 7;
        int l0 = ((cch >> 3) << 4) + n0;   // lane index for (c, n0)
        float m = -3.402823466e38f;
        int wbase = wh * 4;
#pragma unroll
        for (int j = 0; j < 4; ++j) {      // max over (dd,hh) waves and w-pair
            float v0 = sScr[wbase + j][r][l0];
            float v1 = sScr[wbase + j][r][l0 + 1];
            m = fmaxf(m, fmaxf(v0, v1));
        }
        val = m;
    }
    // block sum (wave32 shuffle tree + LDS)
    for (int off = 16; off > 0; off >>= 1) val += __shfl_xor(val, off, 32);
    if (lane == 0) sSum[wv] = val;
    __syncthreads();
    if (tid == 0) {
        float t = 0.0f;
#pragma unroll
        for (int j = 0; j < 8; ++j) t += sSum[j];
        if (partial) partial[(size_t)b * NBLK_PER_B + pd * 15 + ph] = t;
        else         atomicAdd(&accum[b], t);
    }
}

__global__ void zero_kernel(float* p) { p[threadIdx.x] = 0.0f; }

// out[b] = rawsum[b]/(2*1575) + sum_c(conv_bias[c]/2 + bias[c])
__global__ void finalize_kernel(const float* __restrict__ cb,
                                const float* __restrict__ bias,
                                const float* __restrict__ partial, // nullptr => out preaccumulated
                                float* __restrict__ out)
{
    int b = threadIdx.x;   // 128 threads
    float k = 0.0f;
#pragma unroll
    for (int c = 0; c < 16; ++c) k += cb[c] * 0.5f + bias[c];
    float raw;
    if (partial) {
        raw = 0.0f;
        const float* p = partial + (size_t)b * NBLK_PER_B;
        for (int i = 0; i < NBLK_PER_B; ++i) raw += p[i];
    } else {
        raw = out[b];
    }
    out[b] = raw * (1.0f / (2.0f * POOL_CNT)) + k;
}

extern "C" void kernel_launch(void* const* d_in, const int* in_sizes, int n_in,
                              void* d_out, int out_size, void* d_ws, size_t ws_size,
                              hipStream_t stream) {
    const float* x    = (const float*)d_in[0];  // [128,8,16,32,32]
    const float* w    = (const float*)d_in[1];  // [16,8,3,3,3]
    const float* cb   = (const float*)d_in[2];  // [16]
    const float* bias = (const float*)d_in[3];  // [16,1,1,1]
    float* out = (float*)d_out;                 // [128]

    const size_t need = (size_t)128 * NBLK_PER_B * sizeof(float);
    float* partial = (d_ws && ws_size >= need) ? (float*)d_ws : nullptr;

    if (!partial) {
        zero_kernel<<<dim3(1), dim3(128), 0, stream>>>(out);
    }
    conv3d_wmma_pool_kernel<<<dim3(15, 7, 128), dim3(256), 0, stream>>>(
        x, w, partial, partial ? nullptr : out);
    finalize_kernel<<<dim3(1), dim3(128), 0, stream>>>(cb, bias, partial, out);
}